// BalancedGCN_49855980372167
// MI455X (gfx1250) — compile-verified
//
#include <hip/hip_runtime.h>

typedef __attribute__((ext_vector_type(2))) float v2f;
typedef __attribute__((ext_vector_type(8))) float v8f;

#define F_IN  128
#define F_HID 64
#define F_OUT 40

// ---------- graph preprocessing ----------

__global__ void deg_init_kernel(float* __restrict__ deg, int n) {
    int i = blockIdx.x * blockDim.x + threadIdx.x;
    if (i < n) deg[i] = 1.0f;  // self-loop contributes 1 to every node's degree
}

__global__ void deg_accum_kernel(const int* __restrict__ ei, float* __restrict__ deg, int E) {
    int e = blockIdx.x * blockDim.x + threadIdx.x;
    if (e < E) atomicAdd(&deg[ei[E + e]], 1.0f);   // dst row of edge_index
}

__global__ void dinv_kernel(float* __restrict__ deg, int n) {
    int i = blockIdx.x * blockDim.x + threadIdx.x;
    if (i < n) { float d = deg[i]; deg[i] = (d > 0.0f) ? rsqrtf(d) : 0.0f; }
}

__global__ void norm_kernel(const int* __restrict__ ei, const float* __restrict__ dinv,
                            float* __restrict__ norm, int E) {
    int e = blockIdx.x * blockDim.x + threadIdx.x;
    if (e < E) norm[e] = dinv[ei[e]] * dinv[ei[E + e]];
}

// ---------- dense projection via V_WMMA_F32_16X16X4_F32 ----------
// Each wave32 computes one 16x16 output tile; waves within a block cover the
// column tiles of the same 16-row stripe (A rows shared through L0/L2).
// A (16x4 f32): lane l in [0,16) holds row l; VGPR0/1 = K={0,1} lanes 0-15,
//               K={2,3} lanes 16-31.
// B (4x16 f32): lane l holds column l; same K split across halves.
// C/D (16x16):  VGPR v holds M=v (lanes 0-15) / M=v+8 (lanes 16-31), N=lane%16.
// Out-of-range columns (layer 3 only) are loaded with a clamped address —
// D[m][n] depends only on B column n, so garbage columns never contaminate
// stored outputs (store is guarded).
template <bool RELU_IN>
__global__ __launch_bounds__(128) void gemm_wmma_kernel(const float* __restrict__ X,
                                                        const float* __restrict__ W,
                                                        float* __restrict__ H,
                                                        int K, int Fout) {
    const int lane = threadIdx.x & 31;
    const int wave = threadIdx.x >> 5;       // column-tile index within block
    const int half = lane >> 4;              // 0: lanes 0-15, 1: lanes 16-31
    const int l    = lane & 15;
    const int row0 = blockIdx.x * 16;
    const int col  = wave * 16 + l;
    const bool colOK  = (col < Fout);
    const int  colLd  = colOK ? col : (Fout - 1);   // branch-free safe address

    const float* __restrict__ xrow = X + (size_t)(row0 + l) * K + 2 * half;
    const float* __restrict__ wcol = W + colLd + (size_t)(2 * half) * Fout;

    v8f c = {};
#pragma unroll 4
    for (int k0 = 0; k0 < K; k0 += 4) {
        float a0 = xrow[k0];
        float a1 = xrow[k0 + 1];
        if (RELU_IN) { a0 = fmaxf(a0, 0.0f); a1 = fmaxf(a1, 0.0f); }
        v2f a; a.x = a0; a.y = a1;
        v2f b;
        b.x = wcol[(size_t)k0 * Fout];
        b.y = wcol[(size_t)(k0 + 1) * Fout];
        // 8 args: (neg_a, A, neg_b, B, c_mod, C, reuse_a, reuse_b)
        c = __builtin_amdgcn_wmma_f32_16x16x4_f32(false, a, false, b, (short)0, c,
                                                  false, false);
    }

    if (colOK) {
#pragma unroll
        for (int v = 0; v < 8; ++v) {
            const int m = v + 8 * half;
            H[(size_t)(row0 + m) * Fout + col] = c[v];
        }
    }
}

// ---------- aggregation: out = bias + self_loop_term + sum_edges ----------

__global__ void agg_init_kernel(const float* __restrict__ H, const float* __restrict__ dinv,
                                const float* __restrict__ bias, float* __restrict__ A,
                                int N, int F) {
    int idx = blockIdx.x * blockDim.x + threadIdx.x;
    if (idx < N * F) {
        int i = idx / F;
        int f = idx - i * F;
        float di = dinv[i];
        A[idx] = bias[f] + H[idx] * (di * di);   // self-loop norm = dinv[i]^2
    }
}

__global__ void agg_edges_kernel(const float* __restrict__ H, const int* __restrict__ ei,
                                 const float* __restrict__ norm, float* __restrict__ A,
                                 int E, int F) {
    const int chunks = F >> 2;                       // float4 chunks per row
    long long tid   = (long long)blockIdx.x * blockDim.x + threadIdx.x;
    long long total = (long long)E * chunks;
    if (tid >= total) return;
    int e  = (int)(tid / chunks);
    int f0 = (int)(tid - (long long)e * chunks) * 4;

    int   s  = ei[e];
    int   d  = ei[E + e];
    float nv = norm[e];

    const float4 hv = *(const float4*)(H + (size_t)s * F + f0);
    float* __restrict__ out = A + (size_t)d * F + f0;
    atomicAdd(out + 0, hv.x * nv);
    atomicAdd(out + 1, hv.y * nv);
    atomicAdd(out + 2, hv.z * nv);
    atomicAdd(out + 3, hv.w * nv);
}

// ---------- driver ----------

extern "C" void kernel_launch(void* const* d_in, const int* in_sizes, int n_in,
                              void* d_out, int out_size, void* d_ws, size_t ws_size,
                              hipStream_t stream) {
    const float* x  = (const float*)d_in[0];
    const int*   ei = (const int*)d_in[1];   // [2, E] flat: row 0 = src, row 1 = dst
    const float* W1 = (const float*)d_in[2];
    const float* b1 = (const float*)d_in[3];
    const float* W2 = (const float*)d_in[4];
    const float* b2 = (const float*)d_in[5];
    const float* W3 = (const float*)d_in[6];
    const float* b3 = (const float*)d_in[7];
    float* out = (float*)d_out;

    const int N = in_sizes[0] / F_IN;
    const int E = in_sizes[1] / 2;

    // workspace: dinv[N] | norm[E] | bufH[N*64] | bufA[N*64]  (~29 MB)
    float* dinv = (float*)d_ws;
    float* norm = dinv + N;
    float* bufH = norm + E;
    float* bufA = bufH + (size_t)N * F_HID;

    const int T = 256;
    const int gN  = (N + T - 1) / T;
    const int gE  = (E + T - 1) / T;
    const int gNF = (N * F_HID + T - 1) / T;
    const int gNO = (N * F_OUT + T - 1) / T;
    const unsigned gEF = (unsigned)(((long long)E * (F_HID / 4) + T - 1) / T);
    const unsigned gEO = (unsigned)(((long long)E * (F_OUT / 4) + T - 1) / T);

    // --- symmetric GCN normalization ---
    deg_init_kernel <<<gN, T, 0, stream>>>(dinv, N);
    deg_accum_kernel<<<gE, T, 0, stream>>>(ei, dinv, E);
    dinv_kernel     <<<gN, T, 0, stream>>>(dinv, N);
    norm_kernel     <<<gE, T, 0, stream>>>(ei, dinv, norm, E);

    const int rowTiles = (N + 15) / 16;                     // 3125
    const int wavesHid = (F_HID + 15) / 16;                 // 4 column tiles
    const int wavesOut = (F_OUT + 15) / 16;                 // 3 column tiles

    // --- layer 1: H = X @ W1 ; A = agg(H) + b1 ; relu fused into next gemm ---
    gemm_wmma_kernel<false><<<rowTiles, 32 * wavesHid, 0, stream>>>(x, W1, bufH, F_IN, F_HID);
    agg_init_kernel <<<gNF, T, 0, stream>>>(bufH, dinv, b1, bufA, N, F_HID);
    agg_edges_kernel<<<gEF, T, 0, stream>>>(bufH, ei, norm, bufA, E, F_HID);

    // --- layer 2: H = relu(A) @ W2 ; A = agg(H) + b2 ---
    gemm_wmma_kernel<true> <<<rowTiles, 32 * wavesHid, 0, stream>>>(bufA, W2, bufH, F_HID, F_HID);
    agg_init_kernel <<<gNF, T, 0, stream>>>(bufH, dinv, b2, bufA, N, F_HID);
    agg_edges_kernel<<<gEF, T, 0, stream>>>(bufH, ei, norm, bufA, E, F_HID);

    // --- layer 3: H = relu(A) @ W3 ; out = agg(H) + b3 ---
    gemm_wmma_kernel<true> <<<rowTiles, 32 * wavesOut, 0, stream>>>(bufA, W3, bufH, F_HID, F_OUT);
    agg_init_kernel <<<gNO, T, 0, stream>>>(bufH, dinv, b3, out, N, F_OUT);
    agg_edges_kernel<<<gEO, T, 0, stream>>>(bufH, ei, norm, out, E, F_OUT);
}